// SubTokenEmbedding_17403207483995
// MI455X (gfx1250) — compile-verified
//
#include <hip/hip_runtime.h>
#include <stdint.h>

// SubTokenEmbedding: gather(emb_table, subtoken_ids) + sorted-segment sum.
// Memory-bound (~300 MB HBM traffic, ~0 FLOPs/byte) -> CDNA5 async
// global->LDS DMA path (GLOBAL_LOAD_ASYNC_TO_LDS_B128 + ASYNCcnt) for the
// row gather; WMMA is intentionally unused (no matmul structure to exploit).

#define H_EMB     128
#define ROW_BYTES 512                 // 128 * sizeof(float)
#define WPB       8                   // waves per block (wave32)
#define BLOCK_T   (WPB * 32)
#define DEPTH     8                   // async row buffers per wave

__device__ __forceinline__ int lower_bound_dev(const int* __restrict__ a,
                                               int n, int key) {
    int lo = 0, hi = n;
    while (lo < hi) {
        int mid = (int)(((unsigned)lo + (unsigned)hi) >> 1);
        if (a[mid] < key) lo = mid + 1; else hi = mid;
    }
    return lo;
}

__global__ __launch_bounds__(256)
void offsets_kernel(const int* __restrict__ seg, int total, int n_nodes,
                    int* __restrict__ offs) {
    int n = blockIdx.x * blockDim.x + threadIdx.x;
    if (n <= n_nodes) offs[n] = lower_bound_dev(seg, total, n);
}

template <bool USE_OFFS>
__global__ __launch_bounds__(BLOCK_T)
void segsum_kernel(const float* __restrict__ emb,   // [VOCAB, 128]
                   const int*   __restrict__ ids,   // [T]
                   const int*   __restrict__ seg,   // [T] (fallback path only)
                   const int*   __restrict__ offs,  // [N+1] (USE_OFFS path)
                   float*       __restrict__ out,   // [N, 128]
                   int n_nodes, int total) {
    __shared__ float smem[WPB * DEPTH * H_EMB];     // 32 KB / block

    const int wave = (int)(threadIdx.x >> 5);
    const int lane = (int)(threadIdx.x & 31);
    const int node = (int)blockIdx.x * WPB + wave;  // one wave32 per node
    if (node >= n_nodes) return;                    // wave-uniform exit

    int start, end;
    if (USE_OFFS) {
        start = offs[node];
        end   = offs[node + 1];
    } else {
        // even lanes search node, odd lanes search node+1; broadcast results
        int r = lower_bound_dev(seg, total, node + (lane & 1));
        start = __shfl(r, 0, 32);
        end   = __shfl(r, 1, 32);
    }

    // LDS byte offset of this wave's buffers (low 32 bits of the flat address
    // of a __shared__ object == group-relative LDS offset, which is exactly
    // what the async-load VDST operand expects).
    const uint32_t lds_wave = (uint32_t)(uintptr_t)(void*)&smem[wave * (DEPTH * H_EMB)];
    const uint32_t lds_lane = lds_wave + (uint32_t)lane * 16u;
    const size_t   lane_off = (size_t)(lane * 16);   // byte lane within a row

    float4 acc; acc.x = acc.y = acc.z = acc.w = 0.0f;

    for (int t = start; t < end; t += DEPTH) {
        int m = end - t; if (m > DEPTH) m = DEPTH;

        // WAR: prior chunk's ds_load reads of these buffers must complete
        // before the async engine (unordered vs. DS) rewrites them.
        asm volatile("s_wait_dscnt 0" ::: "memory");

        for (int j = 0; j < m; ++j) {
            // Row index is wave-uniform: force it scalar so the row base is
            // computed on the SALU; only "+ lane*16" stays per-lane.
            int id = __builtin_amdgcn_readfirstlane(ids[t + j]);
            // lane L moves bytes [16L, 16L+16) of the 512-B row: one async
            // instruction per wave streams a full embedding row into LDS.
            const char* ga = (const char*)emb
                           + (size_t)(unsigned)id * ROW_BYTES + lane_off;
            uint32_t la = lds_lane + (uint32_t)(j * ROW_BYTES);
            asm volatile("global_load_async_to_lds_b128 %0, %1, off"
                         :: "v"(la), "v"(ga)
                         : "memory");
        }

        asm volatile("s_wait_asynccnt 0" ::: "memory");

        for (int j = 0; j < m; ++j) {
            const float4 v = *reinterpret_cast<const float4*>(
                &smem[wave * (DEPTH * H_EMB) + j * H_EMB + lane * 4]);
            acc.x += v.x; acc.y += v.y; acc.z += v.z; acc.w += v.w;
        }
    }

    // Every node (including empty segments) writes its full row -> zeros OK.
    float4* o = reinterpret_cast<float4*>(out + (size_t)node * H_EMB) + lane;
    *o = acc;
}

extern "C" void kernel_launch(void* const* d_in, const int* in_sizes, int n_in,
                              void* d_out, int out_size, void* d_ws, size_t ws_size,
                              hipStream_t stream) {
    const float* emb = (const float*)d_in[0];   // emb_table    [VOCAB,128] f32
    const int*   ids = (const int*)d_in[1];     // subtoken_ids [T] i32
    const int*   seg = (const int*)d_in[2];     // segment_ids  [T] i32 (sorted)
    // d_in[3] is n_nodes on device; derive it host-side instead:
    const int total   = in_sizes[1];
    const int n_nodes = out_size / H_EMB;
    float* out = (float*)d_out;

    const int nblk = (n_nodes + WPB - 1) / WPB;
    const size_t need = (size_t)(n_nodes + 1) * sizeof(int);

    if (ws_size >= need) {
        int* offs = (int*)d_ws;
        const int nthr = n_nodes + 1;
        offsets_kernel<<<(nthr + 255) / 256, 256, 0, stream>>>(seg, total, n_nodes, offs);
        segsum_kernel<true><<<nblk, BLOCK_T, 0, stream>>>(
            emb, ids, seg, offs, out, n_nodes, total);
    } else {
        segsum_kernel<false><<<nblk, BLOCK_T, 0, stream>>>(
            emb, ids, seg, (const int*)nullptr, out, n_nodes, total);
    }
}